// lovasz_softmax_75402445849362
// MI455X (gfx1250) — compile-verified
//
#include <hip/hip_runtime.h>

// Lovász-Softmax loss, exact sort-free reformulation:
//   loss_c = (sum of positive fg errors)/n_c + (sum of negative fg errors)/P
// (background errors are exactly +-0.0, so the descending sort partitions into
//  [fg e>0][all zeros][fg e<0]; Lovász gradient is 1/n_c resp. 1/P in the
//  non-zero blocks). Whole op becomes an HBM-bound streaming reduction.

#define WAVES_PER_BLOCK 8
#define BLOCK_THREADS   256
#define CHUNK           128      // points per wave per tile (32 lanes x 4)
#define GRID_BLOCKS     1024
#define K_EPS           1e-8f

typedef int v4i __attribute__((vector_size(16)));
typedef __attribute__((address_space(1))) v4i g_v4i;   // global (AS1)
typedef __attribute__((address_space(3))) v4i l_v4i;   // LDS (AS3)

__device__ __forceinline__ void async_copy16(const void* gsrc, void* ldst) {
  // gfx1250: GLOBAL_LOAD_ASYNC_TO_LDS_B128 (tracked with ASYNCcnt)
  __builtin_amdgcn_global_load_async_to_lds_b128(
      (g_v4i*)gsrc, (l_v4i*)ldst, /*offset=*/0, /*cpol=*/0);
}

__device__ __forceinline__ float wave_sum_f(float v) {
#pragma unroll
  for (int off = 16; off > 0; off >>= 1) v += __shfl_xor(v, off, 32);
  return v;
}
__device__ __forceinline__ unsigned wave_sum_u(unsigned v) {
#pragma unroll
  for (int off = 16; off > 0; off >>= 1) v += (unsigned)__shfl_xor((int)v, off, 32);
  return v;
}

__global__ __launch_bounds__(BLOCK_THREADS)
void lovasz_stream_kernel(const float* __restrict__ probas,
                          const int*   __restrict__ labels,
                          float* __restrict__ acc_f,       // [6]: S+0..2, S-0..2
                          unsigned* __restrict__ acc_n,    // [3]: n_c
                          long long P) {
  // [buf][wave][plane][point]: planes = prob0, prob1, prob2, lab1, lab2
  __shared__ __align__(16) float smem[2][WAVES_PER_BLOCK][5][CHUNK];

  const int wave = threadIdx.x >> 5;
  const int lane = threadIdx.x & 31;

  const long long nchunks = P / CHUNK;
  const long long wstride = (long long)gridDim.x * WAVES_PER_BLOCK;
  const long long wid     = (long long)blockIdx.x * WAVES_PER_BLOCK + wave;

  float sp0 = 0.f, sp1 = 0.f, sp2 = 0.f;   // positive-error sums
  float sn0 = 0.f, sn1 = 0.f, sn2 = 0.f;   // negative-error sums
  unsigned c0 = 0u, c1 = 0u, c2 = 0u;      // fg counts

  const int lidx = lane * 4;

  // ---- prologue: issue first tile's async loads ----
  int buf = 0;
  long long c = wid;
  if (c < nchunks) {
    const long long base = c * CHUNK + lidx;
    async_copy16(probas + base,         &smem[0][wave][0][lidx]);
    async_copy16(probas + base + P,     &smem[0][wave][1][lidx]);
    async_copy16(probas + base + 2 * P, &smem[0][wave][2][lidx]);
    async_copy16(labels + base + P,     &smem[0][wave][3][lidx]);
    async_copy16(labels + base + 2 * P, &smem[0][wave][4][lidx]);
  }

  // ---- double-buffered stream: issue tile k+1, wait tile k, process tile k ----
  while (c < nchunks) {
    const long long nx = c + wstride;
    if (nx < nchunks) {
      const int nb = buf ^ 1;
      const long long base = nx * CHUNK + lidx;
      async_copy16(probas + base,         &smem[nb][wave][0][lidx]);
      async_copy16(probas + base + P,     &smem[nb][wave][1][lidx]);
      async_copy16(probas + base + 2 * P, &smem[nb][wave][2][lidx]);
      async_copy16(labels + base + P,     &smem[nb][wave][3][lidx]);
      async_copy16(labels + base + 2 * P, &smem[nb][wave][4][lidx]);
      asm volatile("s_wait_asynccnt 5" ::: "memory");  // tile k complete
    } else {
      asm volatile("s_wait_asynccnt 0" ::: "memory");  // drain
    }

    const float* s0 = smem[buf][wave][0];
    const float* s1 = smem[buf][wave][1];
    const float* s2 = smem[buf][wave][2];
    const float* s3 = smem[buf][wave][3];
    const float* s4 = smem[buf][wave][4];

#pragma unroll
    for (int j = 0; j < 4; ++j) {
      const int idx = lidx + j;
      const float x0 = s0[idx], x1 = s1[idx], x2 = s2[idx];
      const int l1 = __float_as_int(s3[idx]);   // raw int bits via async copy
      const int l2 = __float_as_int(s4[idx]);
      const int lab = l1 + (l2 << 1);

      const float m  = fmaxf(x0, fmaxf(x1, x2));
      const float e0 = __expf(x0 - m), e1 = __expf(x1 - m), e2 = __expf(x2 - m);
      const float inv = 1.0f / (e0 + e1 + e2);
      const float pl  = (lab == 0 ? e0 : (lab == 1 ? e1 : e2)) * inv;
      const float err = -__logf(pl + K_EPS);
      const float ep  = err > 0.0f ? err : 0.0f;
      const float en  = err < 0.0f ? err : 0.0f;

      sp0 += (lab == 0) ? ep : 0.0f;  sn0 += (lab == 0) ? en : 0.0f;
      sp1 += (lab == 1) ? ep : 0.0f;  sn1 += (lab == 1) ? en : 0.0f;
      sp2 += (lab == 2) ? ep : 0.0f;  sn2 += (lab == 2) ? en : 0.0f;
      c0  += (lab == 0) ? 1u : 0u;
      c1  += (lab == 1) ? 1u : 0u;
      c2  += (lab == 2) ? 1u : 0u;
    }

    buf ^= 1;
    c = nx;
  }

  // ---- scalar tail (P % CHUNK points), handled by global wave 0 ----
  if (wid == 0) {
    for (long long i = nchunks * CHUNK + lane; i < P; i += 32) {
      const float x0 = probas[i], x1 = probas[i + P], x2 = probas[i + 2 * P];
      const int lab = labels[i + P] + (labels[i + 2 * P] << 1);
      const float m  = fmaxf(x0, fmaxf(x1, x2));
      const float e0 = __expf(x0 - m), e1 = __expf(x1 - m), e2 = __expf(x2 - m);
      const float inv = 1.0f / (e0 + e1 + e2);
      const float pl  = (lab == 0 ? e0 : (lab == 1 ? e1 : e2)) * inv;
      const float err = -__logf(pl + K_EPS);
      const float ep  = err > 0.0f ? err : 0.0f;
      const float en  = err < 0.0f ? err : 0.0f;
      if (lab == 0)      { sp0 += ep; sn0 += en; c0++; }
      else if (lab == 1) { sp1 += ep; sn1 += en; c1++; }
      else               { sp2 += ep; sn2 += en; c2++; }
    }
  }

  // ---- wave32 reduction, then 9 global atomics per wave ----
  sp0 = wave_sum_f(sp0); sp1 = wave_sum_f(sp1); sp2 = wave_sum_f(sp2);
  sn0 = wave_sum_f(sn0); sn1 = wave_sum_f(sn1); sn2 = wave_sum_f(sn2);
  c0 = wave_sum_u(c0);   c1 = wave_sum_u(c1);   c2 = wave_sum_u(c2);
  if (lane == 0) {
    atomicAdd(&acc_f[0], sp0); atomicAdd(&acc_f[1], sp1); atomicAdd(&acc_f[2], sp2);
    atomicAdd(&acc_f[3], sn0); atomicAdd(&acc_f[4], sn1); atomicAdd(&acc_f[5], sn2);
    atomicAdd(&acc_n[0], c0);  atomicAdd(&acc_n[1], c1);  atomicAdd(&acc_n[2], c2);
  }
}

__global__ void lovasz_init_kernel(float* acc_f, unsigned* acc_n) {
  const int t = threadIdx.x;
  if (t < 6) acc_f[t] = 0.0f;
  if (t < 3) acc_n[t] = 0u;
}

__global__ void lovasz_finalize_kernel(const float* __restrict__ acc_f,
                                       const unsigned* __restrict__ acc_n,
                                       const float* __restrict__ weight,
                                       float* __restrict__ out, float Pf) {
  if (threadIdx.x == 0 && blockIdx.x == 0) {
    float total = 0.0f, count = 0.0f;
#pragma unroll
    for (int cc = 0; cc < 3; ++cc) {
      const unsigned n = acc_n[cc];
      if (n > 0u) {
        const float loss_c = acc_f[cc] / (float)n + acc_f[3 + cc] / Pf;
        total += weight[cc] * loss_c;
        count += 1.0f;
      }
    }
    out[0] = (count > 0.0f) ? (total / count) : 0.0f;
  }
}

extern "C" void kernel_launch(void* const* d_in, const int* in_sizes, int n_in,
                              void* d_out, int out_size, void* d_ws, size_t ws_size,
                              hipStream_t stream) {
  const float* probas = (const float*)d_in[0];
  const float* weight = (const float*)d_in[1];
  const int*   labels = (const int*)d_in[2];
  const long long P = (long long)in_sizes[0] / 3;   // points per class plane

  float*    acc_f = (float*)d_ws;        // 6 floats
  unsigned* acc_n = (unsigned*)(acc_f + 8);  // 3 uints (padded offset)

  lovasz_init_kernel<<<1, 32, 0, stream>>>(acc_f, acc_n);
  lovasz_stream_kernel<<<GRID_BLOCKS, BLOCK_THREADS, 0, stream>>>(
      probas, labels, acc_f, acc_n, P);
  lovasz_finalize_kernel<<<1, 1, 0, stream>>>(acc_f, acc_n, weight,
                                              (float*)d_out, (float)P);
}